// simple_attn_43817256354277
// MI455X (gfx1250) — compile-verified
//
#include <hip/hip_runtime.h>

#define HEADS 8
#define DHEAD 32
#define NROWS 4096
#define CDIM  256
#define LSTR  41   // LDS row stride in floats: odd -> lane-transposed LN reads conflict-free;
                   // 2*41 mod 64 = 18 -> half-wave WMMA fragment rows hit disjoint bank ranges.
#define CHUNK1 128 // rows staged per iteration in pass 1
#define CHUNK2 128 // rows per workgroup in pass 2
#define SPLIT  4   // N-split per (b,h) in pass 1 (partial kv tiles, reduced in pass 2)

typedef __attribute__((ext_vector_type(2))) float v2f;
typedef __attribute__((ext_vector_type(8))) float v8f;

__device__ __forceinline__ v8f wmma_f32_16x16x4(v2f a, v2f b, v8f c) {
  // 8 args: (neg_a, A, neg_b, B, c_mod, C, reuse_a, reuse_b)
  return __builtin_amdgcn_wmma_f32_16x16x4_f32(false, a, false, b, (short)0, c,
                                               false, false);
}

// Async global -> LDS copy (4 bytes/lane), tracked by ASYNCcnt. Falls back to a
// synchronous copy if the builtin is unavailable on this toolchain.
__device__ __forceinline__ void async_copy_f32(const float* g, float* l) {
#if __has_builtin(__builtin_amdgcn_global_load_async_to_lds_b32)
  __builtin_amdgcn_global_load_async_to_lds_b32(
      (__attribute__((address_space(1))) int*)g,
      (__attribute__((address_space(3))) int*)l, 0, 0);
#else
  *l = *g;
#endif
}

template <int N>
__device__ __forceinline__ void wait_async() {
#if __has_builtin(__builtin_amdgcn_s_wait_asynccnt)
  __builtin_amdgcn_s_wait_asynccnt(N);
#else
  asm volatile("s_wait_asynccnt %0" ::"i"(N) : "memory");
#endif
}

// ---------------------------------------------------------------------------
// Pass 1: partial kv[32][32] = (LN_k(x)^T LN_v(x)) / N over an N/SPLIT segment.
// grid = B*HEADS*SPLIT blocks, 128 threads (4 waves). Each wave owns one
// 16x16 tile. Staging is async double-buffered: chunk c+1 streams into LDS
// while LN + WMMA run on chunk c.
// ---------------------------------------------------------------------------
__global__ __launch_bounds__(128)
void kv_pass1(const float* __restrict__ x,
              const float* __restrict__ kw, const float* __restrict__ kb,
              const float* __restrict__ vw, const float* __restrict__ vb,
              float* __restrict__ kvout, int B) {
  __shared__ float xs[2][CHUNK1 * LSTR];
  __shared__ float ks[CHUNK1 * LSTR];
  __shared__ float vs[CHUNK1 * LSTR];
  __shared__ float kwS[DHEAD], kbS[DHEAD], vwS[DHEAD], vbS[DHEAD];

  const int blk  = blockIdx.x;
  const int bh   = blk / SPLIT;
  const int seg  = blk % SPLIT;
  const int b    = bh / HEADS;
  const int h    = bh % HEADS;
  const int lane = threadIdx.x & 31;
  const int wid  = threadIdx.x >> 5;   // 0..3
  const int half = lane >> 4;
  const int l15  = lane & 15;

  if (threadIdx.x < DHEAD) {
    kwS[threadIdx.x] = kw[h * DHEAD + threadIdx.x];
    kbS[threadIdx.x] = kb[h * DHEAD + threadIdx.x];
    vwS[threadIdx.x] = vw[h * DHEAD + threadIdx.x];
    vbS[threadIdx.x] = vb[h * DHEAD + threadIdx.x];
  }

  const int dt = (wid >> 1) * 16;      // tile row offset in kv
  const int et = (wid & 1) * 16;       // tile col offset in kv
  v8f acc = {0.f, 0.f, 0.f, 0.f, 0.f, 0.f, 0.f, 0.f};

  const size_t rowstride = (size_t)CDIM;
  const size_t base = ((size_t)b * NROWS) * rowstride + (size_t)h * DHEAD;
  const int nbeg = seg * (NROWS / SPLIT);
  const int NCH  = (NROWS / SPLIT) / CHUNK1;   // 8 chunks per segment

  auto issue = [&](int c) {
    const int n0 = nbeg + c * CHUNK1;
    float* dst = xs[c & 1];
    #pragma unroll 4
    for (int i = 0; i < CHUNK1 / 4; ++i) {
      int r = wid * (CHUNK1 / 4) + i;
      async_copy_f32(x + base + (size_t)(n0 + r) * rowstride + lane,
                     dst + r * LSTR + lane);
    }
  };

  issue(0);
  for (int c = 0; c < NCH; ++c) {
    wait_async<0>();       // chunk c resident in LDS (per-wave)
    __syncthreads();       // all waves' async writes visible
    if (c + 1 < NCH) issue(c + 1);   // prefetch next chunk into other buffer

    // LayerNorm: each lane owns one row (serial over D, no shuffles).
    {
      int r = wid * 32 + lane;
      const float* row = &xs[c & 1][r * LSTR];
      float s = 0.f, ss = 0.f;
      #pragma unroll
      for (int d = 0; d < DHEAD; ++d) { float t = row[d]; s += t; ss += t * t; }
      float m   = s * (1.0f / DHEAD);
      float var = fmaxf(ss - (float)DHEAD * m * m, 0.0f);       // sum((x-m)^2)
      float inv = 1.0f / (sqrtf(var * (1.0f / (DHEAD - 1))) + 1e-5f);
      #pragma unroll
      for (int d = 0; d < DHEAD; ++d) {
        float t = (row[d] - m) * inv;
        ks[r * LSTR + d] = kwS[d] * t + kbS[d];
        vs[r * LSTR + d] = vwS[d] * t + vbS[d];
      }
    }
    __syncthreads();

    // kv tile accumulation: A = k^T (16x4 per step), B = v (4x16 per step)
    #pragma unroll 8
    for (int kk = 0; kk < CHUNK1; kk += 4) {
      v2f a, bb;
      a.x  = ks[(kk + half * 2 + 0) * LSTR + dt + l15];
      a.y  = ks[(kk + half * 2 + 1) * LSTR + dt + l15];
      bb.x = vs[(kk + half * 2 + 0) * LSTR + et + l15];
      bb.y = vs[(kk + half * 2 + 1) * LSTR + et + l15];
      acc = wmma_f32_16x16x4(a, bb, acc);
    }
    __syncthreads();
  }

  // write this wave's partial 16x16 tile of kv, scaled by 1/N
  const float scale = 1.0f / (float)NROWS;
  float* kvb = kvout + (size_t)blk * (DHEAD * DHEAD);
  #pragma unroll
  for (int r = 0; r < 8; ++r) {
    int row = dt + r + half * 8;         // C/D layout: lanes 16-31 hold M=8..15
    kvb[row * DHEAD + et + l15] = acc[r] * scale;
  }
}

// ---------------------------------------------------------------------------
// Pass 2: out = x_head * kv + x (residual).  grid = B*HEADS*(N/128) blocks,
// 256 threads (8 waves); each wave computes a 16x32 output slab via WMMA.
// kv partials from pass 1 are reduced while staging. Output stores are
// non-temporal so the 128 MB of `out` doesn't evict L2-resident x.
// ---------------------------------------------------------------------------
__global__ __launch_bounds__(256)
void out_pass2(const float* __restrict__ x, const float* __restrict__ kv,
               float* __restrict__ out, int B, int chunks) {
  __shared__ float xs[CHUNK2 * LSTR];
  __shared__ float kvs[DHEAD * LSTR];

  const int gid  = blockIdx.x;
  const int bh   = gid / chunks;
  const int c0   = gid % chunks;
  const int b    = bh / HEADS;
  const int h    = bh % HEADS;
  const int lane = threadIdx.x & 31;
  const int wid  = threadIdx.x >> 5;   // 0..7
  const int half = lane >> 4;
  const int l15  = lane & 15;

  const size_t rowstride = (size_t)CDIM;
  const size_t base = ((size_t)b * NROWS) * rowstride + (size_t)h * DHEAD;
  const int n0 = c0 * CHUNK2;

  // async-stage 128 rows of x for this head
  #pragma unroll 4
  for (int i = 0; i < CHUNK2 / 8; ++i) {
    int r = wid * (CHUNK2 / 8) + i;
    async_copy_f32(x + base + (size_t)(n0 + r) * rowstride + lane,
                   xs + r * LSTR + lane);
  }

  // reduce SPLIT partial kv tiles into LDS (tiny, L2-resident, reused 32x)
  for (int e = threadIdx.x; e < DHEAD * DHEAD; e += 256) {
    float s = 0.f;
    #pragma unroll
    for (int sg = 0; sg < SPLIT; ++sg)
      s += kv[((size_t)bh * SPLIT + sg) * (DHEAD * DHEAD) + e];
    kvs[(e >> 5) * LSTR + (e & 31)] = s;
  }

  wait_async<0>();
  __syncthreads();

  v8f acc0 = {0.f, 0.f, 0.f, 0.f, 0.f, 0.f, 0.f, 0.f};
  v8f acc1 = {0.f, 0.f, 0.f, 0.f, 0.f, 0.f, 0.f, 0.f};
  const int rbase = wid * 16;

  #pragma unroll
  for (int kk = 0; kk < DHEAD; kk += 4) {
    v2f a, b0, b1;
    a.x  = xs[(rbase + l15) * LSTR + kk + half * 2 + 0];   // A[M=row, K=d]
    a.y  = xs[(rbase + l15) * LSTR + kk + half * 2 + 1];
    b0.x = kvs[(kk + half * 2 + 0) * LSTR + l15];          // B[K=d, N=e]
    b0.y = kvs[(kk + half * 2 + 1) * LSTR + l15];
    b1.x = kvs[(kk + half * 2 + 0) * LSTR + 16 + l15];
    b1.y = kvs[(kk + half * 2 + 1) * LSTR + 16 + l15];
    acc0 = wmma_f32_16x16x4(a, b0, acc0);
    acc1 = wmma_f32_16x16x4(a, b1, acc1);
  }

  // residual add + non-temporal store (half-wave writes 64B contiguous runs)
  #pragma unroll
  for (int r = 0; r < 8; ++r) {
    int rl = rbase + r + half * 8;
    size_t o = base + (size_t)(n0 + rl) * rowstride;
    __builtin_nontemporal_store(acc0[r] + xs[rl * LSTR + l15],      &out[o + l15]);
    __builtin_nontemporal_store(acc1[r] + xs[rl * LSTR + 16 + l15], &out[o + 16 + l15]);
  }
}

extern "C" void kernel_launch(void* const* d_in, const int* in_sizes, int n_in,
                              void* d_out, int out_size, void* d_ws, size_t ws_size,
                              hipStream_t stream) {
  const float* x  = (const float*)d_in[0];
  const float* kw = (const float*)d_in[1];
  const float* kb = (const float*)d_in[2];
  const float* vw = (const float*)d_in[3];
  const float* vb = (const float*)d_in[4];
  float* out = (float*)d_out;
  float* kvw = (float*)d_ws;   // B*H*SPLIT*32*32 floats = 4 MB of partial kv

  const int B = in_sizes[0] / (NROWS * CDIM);

  kv_pass1<<<B * HEADS * SPLIT, 128, 0, stream>>>(x, kw, kb, vw, vb, kvw, B);

  const int chunks = NROWS / CHUNK2;   // 32
  out_pass2<<<B * HEADS * chunks, 256, 0, stream>>>(x, kvw, out, B, chunks);
}